// SparseScaledDotProductAttention_3582002725047
// MI455X (gfx1250) — compile-verified
//
#include <hip/hip_runtime.h>
#include <hip/hip_bf16.h>
#include <math.h>

// Problem constants from the reference: B=1, H=16, N=2048, D=64, half-window=1.
#define Hh   16
#define Nn   2048
#define Dd   64
#define QB   16          // queries per wave / per WMMA tile

typedef __attribute__((ext_vector_type(2))) float v2f;
typedef __attribute__((ext_vector_type(8))) float v8f;

// ---------------------------------------------------------------------------
// Kernel 1: zero-fill the dense attn region (H*N*N floats = 268 MB).
// This is the bandwidth-bound part of the whole problem (~11.5us @ 23.3 TB/s).
// ---------------------------------------------------------------------------
__global__ __launch_bounds__(256) void band_attn_zero_kernel(float4* __restrict__ p,
                                                             long long n4) {
    long long i = (long long)blockIdx.x * blockDim.x + threadIdx.x;
    if (i < n4) p[i] = make_float4(0.f, 0.f, 0.f, 0.f);
}

// ---------------------------------------------------------------------------
// Kernel 2: one wave (32 lanes) per (head, 16-query block).
//   S = Q_blk * K_blk^T via 16 chained V_WMMA_F32_16X16X4_F32 (fp32 WMMA),
//   two boundary-crossing band scores via shuffle-reduced dot products,
//   per-row 3-way softmax, band writes into attn, out = attn_band * V.
// ---------------------------------------------------------------------------
__global__ __launch_bounds__(32) void band_attn_kernel(const float* __restrict__ Q,
                                                       const float* __restrict__ K,
                                                       const float* __restrict__ V,
                                                       float* __restrict__ out,
                                                       float* __restrict__ attn) {
    const int lane = threadIdx.x;                 // 0..31 (wave32)
    const int blk  = blockIdx.x;                  // 0 .. H*(N/QB)-1
    const int h    = blk / (Nn / QB);
    const int q0   = (blk % (Nn / QB)) * QB;

    const float* Qh = Q + (size_t)h * Nn * Dd;
    const float* Kh = K + (size_t)h * Nn * Dd;
    const float* Vh = V + (size_t)h * Nn * Dd;
    float* outh  = out  + (size_t)h * Nn * Dd;
    float* attnh = attn + (size_t)h * Nn * Nn;

    // 32-bit A 16x4 fragment layout: element [m][k] lives in
    //   lane = m + 16*(k>>1), reg = k&1.
    // B 4x16 is the symmetric dual: element [k][n] lives in
    //   lane = n + 16*(k>>1), reg = k&1.
    // => both Q (as A, rows=queries) and K (as B, cols=keys, B[d][j]=K[j][d])
    //    load with the SAME per-lane addressing.
    const int mrow  = lane & 15;
    const int khalf = lane >> 4;                  // which K-dim pair this lane holds

    v8f acc = {};                                  // 16x16 fp32 score tile
    #pragma unroll
    for (int kk = 0; kk < Dd; kk += 4) {
        v2f a = *(const v2f*)(Qh + (size_t)(q0 + mrow) * Dd + kk + 2 * khalf);
        v2f b = *(const v2f*)(Kh + (size_t)(q0 + mrow) * Dd + kk + 2 * khalf);
        acc = __builtin_amdgcn_wmma_f32_16x16x4_f32(false, a, false, b,
                                                    (short)0, acc, false, false);
    }

    // Two band scores that cross the 16-wide tile boundary:
    //   eL = dot(Q[q0],   K[q0-1]),  eR = dot(Q[q0+15], K[q0+16])
    float pL = 0.f, pR = 0.f;
    {
        const int d0 = lane * 2;                   // 2 dims per lane
        if (q0 > 0) {
            pL = Qh[(size_t)q0 * Dd + d0]     * Kh[(size_t)(q0 - 1) * Dd + d0]
               + Qh[(size_t)q0 * Dd + d0 + 1] * Kh[(size_t)(q0 - 1) * Dd + d0 + 1];
        }
        if (q0 + QB < Nn) {
            pR = Qh[(size_t)(q0 + 15) * Dd + d0]     * Kh[(size_t)(q0 + 16) * Dd + d0]
               + Qh[(size_t)(q0 + 15) * Dd + d0 + 1] * Kh[(size_t)(q0 + 16) * Dd + d0 + 1];
        }
        #pragma unroll
        for (int off = 16; off; off >>= 1) {
            pL += __shfl_xor(pL, off, 32);
            pR += __shfl_xor(pR, off, 32);
        }
    }
    const float eL = pL, eR = pR;

    // Spill score tile to LDS to pull out the 3 diagonals.
    // C/D layout: acc[r] on lane l is S[r + 8*(l>>4)][l&15].
    __shared__ float sS[QB][QB];
    __shared__ float sA[QB][4];                    // softmax weights (ap,ac,an)
    {
        const int n   = lane & 15;
        const int mhi = (lane >> 4) * 8;
        #pragma unroll
        for (int r = 0; r < 8; ++r)
            sS[mhi + r][n] = acc[r];
    }
    __syncthreads();

    // Per-query softmax over the 3 band scores (lanes 0..15, one row each).
    const float scale = 0.125f;                    // 1/sqrt(64)
    if (lane < QB) {
        const int m  = lane;
        const int qg = q0 + m;                     // global query index
        float sp = (m == 0)  ? eL : sS[m][m - 1];
        float sc = sS[m][m];
        float sn = (m == 15) ? eR : sS[m][m + 1];
        if (qg == 0)        sp = -INFINITY;        // band mask at global edges
        if (qg == Nn - 1)   sn = -INFINITY;
        sp *= scale; sc *= scale; sn *= scale;
        const float mx  = fmaxf(sc, fmaxf(sp, sn));
        const float ep  = __expf(sp - mx);         // exp(-inf)=0 handles edges
        const float ec  = __expf(sc - mx);
        const float en  = __expf(sn - mx);
        const float inv = 1.0f / (ep + ec + en);
        const float ap = ep * inv, ac = ec * inv, an = en * inv;
        sA[m][0] = ap; sA[m][1] = ac; sA[m][2] = an;

        // Band entries of attn; everything else was pre-zeroed by kernel 1.
        float* arow = attnh + (size_t)qg * Nn;
        if (qg > 0)      arow[qg - 1] = ap;
        arow[qg] = ac;
        if (qg < Nn - 1) arow[qg + 1] = an;
    }
    __syncthreads();

    // out[q] = ap*V[q-1] + ac*V[q] + an*V[q+1]; 32 lanes x float2 cover D=64.
    const int d0 = lane * 2;
    #pragma unroll 4
    for (int m = 0; m < QB; ++m) {
        const int qg = q0 + m;
        const float ap = sA[m][0], ac = sA[m][1], an = sA[m][2];
        const int jp = (qg > 0)      ? qg - 1 : qg;   // weight is exactly 0 when clamped
        const int jn = (qg < Nn - 1) ? qg + 1 : qg;
        const float2 vp = *(const float2*)(Vh + (size_t)jp * Dd + d0);
        const float2 vc = *(const float2*)(Vh + (size_t)qg * Dd + d0);
        const float2 vn = *(const float2*)(Vh + (size_t)jn * Dd + d0);
        float2 o;
        o.x = ap * vp.x + ac * vc.x + an * vn.x;
        o.y = ap * vp.y + ac * vc.y + an * vn.y;
        *(float2*)(outh + (size_t)qg * Dd + d0) = o;
    }
}

// ---------------------------------------------------------------------------
extern "C" void kernel_launch(void* const* d_in, const int* in_sizes, int n_in,
                              void* d_out, int out_size, void* d_ws, size_t ws_size,
                              hipStream_t stream) {
    const float* Q = (const float*)d_in[0];
    const float* K = (const float*)d_in[1];
    const float* V = (const float*)d_in[2];

    float* out  = (float*)d_out;                       // [1,16,2048,64]
    float* attn = out + (size_t)Hh * Nn * Dd;          // [1,16,2048,2048]

    // 1) zero the dense attn region (268 MB of float4 stores — the real cost)
    const long long n4 = (long long)Hh * Nn * Nn / 4;  // 16,777,216 float4s
    const int zthreads = 256;
    const long long zblocks = (n4 + zthreads - 1) / zthreads;
    band_attn_zero_kernel<<<(unsigned)zblocks, zthreads, 0, stream>>>((float4*)attn, n4);

    // 2) banded attention: one wave per (head, 16-query block)
    const int nblocks = Hh * (Nn / QB);                // 2048 workgroups
    band_attn_kernel<<<nblocks, 32, 0, stream>>>(Q, K, V, out, attn);
}